// GumbelTopK_31920196944434
// MI455X (gfx1250) — compile-verified
//
#include <hip/hip_runtime.h>
#include <hip/hip_bf16.h>

// Gumbel top-K (B=512, N=4096, K=16, tau=2/3) for gfx1250 (MI455X).
// Roofline: 256MB streamed output + 16MB input @ 23.3TB/s => ~11.7us floor;
// compute (33.5M v_exp_f32) is far below any ceiling => pure streaming-store
// problem. CDNA5 paths used: async global->LDS staging (ASYNCcnt) +
// s_wait_asynccnt, wave32 butterfly reductions, NT b128 stores so the 256MB
// never-reread output doesn't thrash the 192MB L2. No matmul in this op =>
// WMMA intentionally unused.

#define BROWS   512
#define NCOLS   4096
#define KTOP    16
#define THREADS 256
#define NWAVES  (THREADS / 32)
#define EPT     16            // elements per thread
#define CHUNKS  4             // float4 chunks per thread

// log2(e) / tau = 1.4426950408889634 * 1.5
#define INV_TAU_LOG2E 2.1640425613334451f
// log(float32 tiny) = log(1.17549435e-38)
#define LOG_EPS (-87.33654785f)

typedef float v4f __attribute__((ext_vector_type(4)));

// Low 32 bits of a generic pointer to LDS == byte offset within the
// wave's LDS allocation (aperture lives in the high half).
__device__ __forceinline__ unsigned lds_off(const void* p) {
  return (unsigned)(unsigned long long)p;
}

__global__ __launch_bounds__(THREADS)
void gumbel_topk_kernel(const float* __restrict__ logits,
                        const float* __restrict__ gumbel,
                        float* __restrict__ out_st,
                        float* __restrict__ out_soft)
{
  __shared__ float    s_log[NCOLS];
  __shared__ float    s_gum[NCOLS];
  __shared__ float    red_v[NWAVES];   // argmax stage: values
  __shared__ unsigned red_i[NWAVES];   // argmax stage: indices
  __shared__ float    red_s[NWAVES];   // sum stage (separate array => 2 barriers/iter)

  const int tid  = threadIdx.x;
  const int b    = blockIdx.x;
  const int lane = tid & 31;
  const int wave = tid >> 5;

  const float* glog = logits + (size_t)b * NCOLS;
  const float* ggum = gumbel + (size_t)b * NCOLS;

  // ---- CDNA5 async global->LDS staging (ASYNCcnt path) ----
  const unsigned lbase_log = lds_off(&s_log[0]);
  const unsigned lbase_gum = lds_off(&s_gum[0]);
#pragma unroll
  for (int c = 0; c < CHUNKS; ++c) {
    unsigned off = (unsigned)((c * 1024 + tid * 4) * 4);  // byte offset
    asm volatile("global_load_async_to_lds_b128 %0, %1, %2 offset:0"
                 :: "v"(lbase_log + off), "v"(off), "s"(glog) : "memory");
    asm volatile("global_load_async_to_lds_b128 %0, %1, %2 offset:0"
                 :: "v"(lbase_gum + off), "v"(off), "s"(ggum) : "memory");
  }
  // Each thread staged exactly the bytes it reads below, so waiting on this
  // wave's own ASYNCcnt is sufficient (no cross-wave LDS consumption here).
  asm volatile("s_wait_asynccnt 0x0" ::: "memory");

  float noisy[EPT];
#pragma unroll
  for (int c = 0; c < CHUNKS; ++c) {
    int base = c * 1024 + tid * 4;
    v4f lv = *reinterpret_cast<const v4f*>(&s_log[base]);  // ds_load_b128
    v4f gv = *reinterpret_cast<const v4f*>(&s_gum[base]);  // ds_load_b128
#pragma unroll
    for (int r = 0; r < 4; ++r)
      noisy[c * 4 + r] = lv[r] + gv[r];
  }
  __syncthreads();

  for (int k = 0; k < KTOP; ++k) {
    // ---- 1) row argmax of noisy (== argmax of softmax), first-index wins ----
    float    mv = noisy[0];
    unsigned mi = (unsigned)(tid * 4);
#pragma unroll
    for (int j = 1; j < EPT; ++j) {
      unsigned gi = (unsigned)((j >> 2) * 1024 + tid * 4 + (j & 3));
      if (noisy[j] > mv) { mv = noisy[j]; mi = gi; }
    }
#pragma unroll
    for (int off = 16; off > 0; off >>= 1) {          // wave32 butterfly
      float    ov = __shfl_xor(mv, off, 32);
      unsigned oi = (unsigned)__shfl_xor((int)mi, off, 32);
      if (ov > mv || (ov == mv && oi < mi)) { mv = ov; mi = oi; }
    }
    if (lane == 0) { red_v[wave] = mv; red_i[wave] = mi; }
    __syncthreads();                                   // barrier 1/2
    // every thread reduces the 8 wave results itself (LDS broadcast reads)
    float    m    = red_v[0];
    unsigned amax = red_i[0];
#pragma unroll
    for (int w = 1; w < NWAVES; ++w) {
      float    wv = red_v[w];
      unsigned wi = red_i[w];
      if (wv > m || (wv == m && wi < amax)) { m = wv; amax = wi; }
    }

    // ---- 2) sum of exp((x - m)/tau)  (v_exp_f32 via exp2) ----
    float e[EPT];
    float s = 0.0f;
#pragma unroll
    for (int j = 0; j < EPT; ++j) {
      e[j] = __builtin_exp2f((noisy[j] - m) * INV_TAU_LOG2E);
      s += e[j];
    }
#pragma unroll
    for (int off = 16; off > 0; off >>= 1)
      s += __shfl_xor(s, off, 32);
    if (lane == 0) red_s[wave] = s;
    __syncthreads();                                   // barrier 2/2
    float tot = 0.0f;
#pragma unroll
    for (int w = 0; w < NWAVES; ++w) tot += red_s[w];
    const float inv = 1.0f / tot;

    // ---- 3) stream soft_k and st_k with non-temporal b128 stores ----
    float* st_row   = out_st   + (size_t)k * BROWS * NCOLS + (size_t)b * NCOLS;
    float* soft_row = out_soft + (size_t)k * BROWS * NCOLS + (size_t)b * NCOLS;
#pragma unroll
    for (int c = 0; c < CHUNKS; ++c) {
      int base = c * 1024 + tid * 4;
      v4f sv, hv;
#pragma unroll
      for (int r = 0; r < 4; ++r) {
        int   j  = c * 4 + r;
        float sf = e[j] * inv;
        sv[r] = sf;
        // straight-through: (1 - s) + s at the hard index, exact 0 elsewhere
        hv[r] = ((unsigned)(base + r) == amax) ? ((1.0f - sf) + sf) : 0.0f;
      }
      __builtin_nontemporal_store(sv, (v4f*)(soft_row + base));
      __builtin_nontemporal_store(hv, (v4f*)(st_row + base));
    }

    // ---- 4) mask the selected element for the next round ----
    // (The selected element is always the current max; masking drops it by
    //  log(FLT_TINY) so the next softmax effectively excludes it.)
#pragma unroll
    for (int j = 0; j < EPT; ++j) {
      unsigned gi = (unsigned)((j >> 2) * 1024 + tid * 4 + (j & 3));
      if (gi == amax) noisy[j] += LOG_EPS;
    }
    // Cross-iteration LDS hazards: red_v/red_i reads happen between barrier 1
    // and barrier 2; they are rewritten only after the *next* iteration's
    // pre-barrier region, which is after barrier 2. red_s reads happen after
    // barrier 2 and it is rewritten only after the next barrier 1. Safe.
  }
}

extern "C" void kernel_launch(void* const* d_in, const int* in_sizes, int n_in,
                              void* d_out, int out_size, void* d_ws, size_t ws_size,
                              hipStream_t stream) {
  const float* logits = (const float*)d_in[0];
  const float* gumbel = (const float*)d_in[1];
  // d_in[2] is k (== 16, compile-time constant here)
  float* out      = (float*)d_out;
  float* out_st   = out;                                   // st:  [K,B,N]
  float* out_soft = out + (size_t)KTOP * BROWS * NCOLS;    // soft:[K,B,N]

  dim3 grid(BROWS), block(THREADS);
  hipLaunchKernelGGL(gumbel_topk_kernel, grid, block, 0, stream,
                     logits, gumbel, out_st, out_soft);
}